// SpGAT_59098749993531
// MI455X (gfx1250) — compile-verified
//
#include <hip/hip_runtime.h>
#include <math.h>

typedef float v2f __attribute__((ext_vector_type(2)));
typedef float v8f __attribute__((ext_vector_type(8)));

#define KB_ALPHA 0.2f

__device__ __forceinline__ float kb_lrelu(float x) { return x > 0.f ? x : KB_ALPHA * x; }
__device__ __forceinline__ float kb_elu(float x)   { return x > 0.f ? x : (__expf(x) - 1.f); }

__device__ __forceinline__ void kb_atomadd(float* p, float v) {
    atomicAdd(p, v);
}

// ---------------------------------------------------------------------------
// WMMA f32 GEMM, K fixed at compile time (fully unrolled), row-major:
//   C[M,N] = A[M,K] @ B[K,N]
// One wave computes a 16(M) x 64(N) tile with 4 accumulators, reusing the A
// fragment 4x per K-step. Requirements: N % 64 == 0. M may be ragged: A rows
// are clamped (garbage accumulates into D rows >= M, which are never stored).
// Fragment layouts per CDNA5 ISA 7.12.2:
//   A lane l: (M = l%16, K = k + 2*(l>=16))   -> a.x = K+0, a.y = K+1
//   B lane l: (K = k + 2*(l>=16), N = l%16)   -> b.x = row K, b.y = row K+1
//   D vgpr v, lane l: (M = v + 8*(l>=16), N = l%16)
// ---------------------------------------------------------------------------
template <int K>
__global__ void gemm_wmma_f32_k(const float* __restrict__ A, const float* __restrict__ B,
                                float* __restrict__ C, int M, int N)
{
    const int lane = threadIdx.x & 31;
    const int wave = threadIdx.x >> 5;
    const int nTilesN = N >> 6;                    // 64-wide N tiles
    const int nTilesM = (M + 15) >> 4;
    const int tile = blockIdx.x * (blockDim.x >> 5) + wave;
    if (tile >= nTilesN * nTilesM) return;         // wave-uniform: EXEC all-ones inside
    const int tm = tile / nTilesN;
    const int tn = tile - tm * nTilesN;
    const int half = lane >> 4;
    const int lm   = lane & 15;
    int row = tm * 16 + lm;
    if (row >= M) row = M - 1;                     // clamp: no divergent load guard
    const int col0 = tn * 64 + lm;
    const float* Ap = A + (size_t)row * K + 2 * half;
    const float* Bp = B + (size_t)(2 * half) * N + col0;
    v8f acc0 = {}, acc1 = {}, acc2 = {}, acc3 = {};
#pragma unroll
    for (int kk = 0; kk < K; kk += 4) {
        v2f a;
        a.x = Ap[kk];
        a.y = Ap[kk + 1];
        const float* Bk = Bp + (size_t)kk * N;
        v2f b0, b1, b2, b3;
        b0.x = Bk[0];   b0.y = Bk[N];
        b1.x = Bk[16];  b1.y = Bk[N + 16];
        b2.x = Bk[32];  b2.y = Bk[N + 32];
        b3.x = Bk[48];  b3.y = Bk[N + 48];
        acc0 = __builtin_amdgcn_wmma_f32_16x16x4_f32(false, a, false, b0, (short)0, acc0, false, false);
        acc1 = __builtin_amdgcn_wmma_f32_16x16x4_f32(false, a, false, b1, (short)0, acc1, false, false);
        acc2 = __builtin_amdgcn_wmma_f32_16x16x4_f32(false, a, false, b2, (short)0, acc2, false, false);
        acc3 = __builtin_amdgcn_wmma_f32_16x16x4_f32(false, a, false, b3, (short)0, acc3, false, false);
    }
#pragma unroll
    for (int v = 0; v < 8; ++v) {
        const int r = tm * 16 + v + 8 * half;
        if (r < M) {
            float* Cr = C + (size_t)r * N + col0;
            Cr[0]  = acc0[v];
            Cr[16] = acc1[v];
            Cr[32] = acc2[v];
            Cr[48] = acc3[v];
        }
    }
}

// ---------------------------------------------------------------------------
// Weight repacking kernels (tiny). a_heads: [2,64,384], a_out: [128,384].
// ---------------------------------------------------------------------------
__global__ void pack_AB1(const float* __restrict__ a_heads, float* __restrict__ AB1) {
    int idx = blockIdx.x * blockDim.x + threadIdx.x;   // 128 x 256
    if (idx >= 128 * 256) return;
    int k = idx >> 8, j = idx & 255;
    int h = j >> 7, jj = j & 127;
    int sd = jj >> 6, r = jj & 63;                      // 0=src block, 1=dst block
    AB1[idx] = a_heads[h * 24576 + r * 384 + sd * 128 + k];
}

__global__ void pack_C1(const float* __restrict__ a_heads, float* __restrict__ C1) {
    int idx = blockIdx.x * blockDim.x + threadIdx.x;   // 128 x 128
    if (idx >= 128 * 128) return;
    int k = idx >> 7, j = idx & 127;
    int h = j >> 6, r = j & 63;
    C1[idx] = a_heads[h * 24576 + r * 384 + 256 + k];
}

__global__ void pack_AB2(const float* __restrict__ a_out, float* __restrict__ AB2) {
    int idx = blockIdx.x * blockDim.x + threadIdx.x;   // 128 x 256
    if (idx >= 128 * 256) return;
    int k = idx >> 8, j = idx & 255;
    int sd = j >> 7, r = j & 127;
    AB2[idx] = a_out[r * 384 + sd * 128 + k];
}

__global__ void pack_Crel2(const float* __restrict__ a_out, float* __restrict__ Cr) {
    int idx = blockIdx.x * blockDim.x + threadIdx.x;   // 128 x 128
    if (idx >= 128 * 128) return;
    int k = idx >> 7, r = idx & 127;
    Cr[idx] = a_out[r * 384 + 256 + k];
}

// ---------------------------------------------------------------------------
// Layer-1 edge kernel: one wave per edge. m[128] = P1[s,srcblk]+P1[d,dstblk]+R,
// dot with a2 per head, w = exp(-leaky_relu(dot)), atomic scatter into H1/rowsum.
// P1 layout: [N,256] = [h0 src | h0 dst | h1 src | h1 dst]. R layout: [*,128].
// ---------------------------------------------------------------------------
__global__ void edge_att1(const int* __restrict__ edge_list,       // [2,E]
                          const int* __restrict__ edge_list_nhop,  // [2,EN]
                          const int* __restrict__ edge_type_nhop,  // [EN,2]
                          const float* __restrict__ P1,
                          const float* __restrict__ R1,            // [E,128]
                          const float* __restrict__ relC,          // [NREL,128]
                          const float* __restrict__ a2,            // [2*64]
                          float* __restrict__ H1,                  // [N,128]
                          float* __restrict__ rowsum1,             // [N,2]
                          int E, int EN)
{
    const int lane = threadIdx.x & 31;
    const long long e = (long long)blockIdx.x * (blockDim.x >> 5) + (threadIdx.x >> 5);
    if (e >= (long long)E + EN) return;
    int s, d;
    float r0, r1, r2, r3;
    if (e < E) {
        s = edge_list[e]; d = edge_list[(size_t)E + e];
        const float* Rp = R1 + (size_t)e * 128;
        r0 = Rp[lane];      r1 = Rp[lane + 32];
        r2 = Rp[lane + 64]; r3 = Rp[lane + 96];
    } else {
        long long i = e - E;
        s = edge_list_nhop[i]; d = edge_list_nhop[(size_t)EN + i];
        int t0 = edge_type_nhop[2 * i], t1 = edge_type_nhop[2 * i + 1];
        const float* c0 = relC + (size_t)t0 * 128;
        const float* c1 = relC + (size_t)t1 * 128;
        r0 = c0[lane]      + c1[lane];
        r1 = c0[lane + 32] + c1[lane + 32];
        r2 = c0[lane + 64] + c1[lane + 64];
        r3 = c0[lane + 96] + c1[lane + 96];
    }
    const float* Ps = P1 + (size_t)s * 256;
    const float* Pd = P1 + (size_t)d * 256;
    float m0a = Ps[lane]        + Pd[64 + lane]   + r0;   // head0 comps [0,32)
    float m0b = Ps[lane + 32]   + Pd[96 + lane]   + r1;   // head0 comps [32,64)
    float m1a = Ps[128 + lane]  + Pd[192 + lane]  + r2;   // head1 comps [0,32)
    float m1b = Ps[160 + lane]  + Pd[224 + lane]  + r3;   // head1 comps [32,64)
    float d0 = a2[lane] * m0a + a2[lane + 32] * m0b;
    float d1 = a2[64 + lane] * m1a + a2[96 + lane] * m1b;
    for (int off = 16; off; off >>= 1) {
        d0 += __shfl_xor(d0, off, 32);
        d1 += __shfl_xor(d1, off, 32);
    }
    float w0 = __expf(-kb_lrelu(d0));
    float w1 = __expf(-kb_lrelu(d1));
    float* Hs = H1 + (size_t)s * 128;
    kb_atomadd(Hs + lane,       w0 * m0a);
    kb_atomadd(Hs + 32 + lane,  w0 * m0b);
    kb_atomadd(Hs + 64 + lane,  w1 * m1a);
    kb_atomadd(Hs + 96 + lane,  w1 * m1b);
    if (lane == 0) {
        kb_atomadd(rowsum1 + 2 * (size_t)s,     w0);
        kb_atomadd(rowsum1 + 2 * (size_t)s + 1, w1);
    }
}

__global__ void node1_finalize(const float* __restrict__ H1,
                               const float* __restrict__ rowsum1,
                               float* __restrict__ x2, int N)
{
    int idx = blockIdx.x * blockDim.x + threadIdx.x;
    if (idx >= N * 128) return;
    int n = idx >> 7, j = idx & 127, h = j >> 6;
    float rs = rowsum1[2 * n + h];
    rs = (rs == 0.f) ? 1e-12f : rs;
    x2[idx] = kb_elu(H1[idx] / rs);
}

// ---------------------------------------------------------------------------
// Layer-2 edge kernel: one wave per edge, 128-wide features.
// P2 layout: [N,256] = [src proj | dst proj].
// ---------------------------------------------------------------------------
__global__ void edge_att2(const int* __restrict__ edge_list,
                          const int* __restrict__ edge_list_nhop,
                          const int* __restrict__ edge_type,       // [E]
                          const int* __restrict__ edge_type_nhop,  // [EN,2]
                          const float* __restrict__ P2,
                          const float* __restrict__ relproj2,      // [NREL,128]
                          const float* __restrict__ a2o,           // [128]
                          float* __restrict__ H2,                  // [N,128]
                          float* __restrict__ rowsum2,             // [N]
                          int E, int EN)
{
    const int lane = threadIdx.x & 31;
    const long long e = (long long)blockIdx.x * (blockDim.x >> 5) + (threadIdx.x >> 5);
    if (e >= (long long)E + EN) return;
    int s, d;
    float r0, r1, r2, r3;
    if (e < E) {
        s = edge_list[e]; d = edge_list[(size_t)E + e];
        const float* rp = relproj2 + (size_t)edge_type[e] * 128;
        r0 = rp[lane];      r1 = rp[32 + lane];
        r2 = rp[64 + lane]; r3 = rp[96 + lane];
    } else {
        long long i = e - E;
        s = edge_list_nhop[i]; d = edge_list_nhop[(size_t)EN + i];
        int t0 = edge_type_nhop[2 * i], t1 = edge_type_nhop[2 * i + 1];
        const float* c0 = relproj2 + (size_t)t0 * 128;
        const float* c1 = relproj2 + (size_t)t1 * 128;
        r0 = c0[lane]      + c1[lane];
        r1 = c0[32 + lane] + c1[32 + lane];
        r2 = c0[64 + lane] + c1[64 + lane];
        r3 = c0[96 + lane] + c1[96 + lane];
    }
    const float* Ps = P2 + (size_t)s * 256;
    const float* Pd = P2 + (size_t)d * 256 + 128;
    float m0 = Ps[lane]      + Pd[lane]      + r0;
    float m1 = Ps[32 + lane] + Pd[32 + lane] + r1;
    float m2 = Ps[64 + lane] + Pd[64 + lane] + r2;
    float m3 = Ps[96 + lane] + Pd[96 + lane] + r3;
    float dot = a2o[lane] * m0 + a2o[32 + lane] * m1
              + a2o[64 + lane] * m2 + a2o[96 + lane] * m3;
    for (int off = 16; off; off >>= 1) dot += __shfl_xor(dot, off, 32);
    float w = __expf(-kb_lrelu(dot));
    float* Hs = H2 + (size_t)s * 128;
    kb_atomadd(Hs + lane,      w * m0);
    kb_atomadd(Hs + 32 + lane, w * m1);
    kb_atomadd(Hs + 64 + lane, w * m2);
    kb_atomadd(Hs + 96 + lane, w * m3);
    if (lane == 0) kb_atomadd(rowsum2 + s, w);
}

__global__ void node2_finalize(const float* __restrict__ H2,
                               const float* __restrict__ rowsum2,
                               float* __restrict__ out, int N)
{
    int idx = blockIdx.x * blockDim.x + threadIdx.x;
    if (idx >= N * 128) return;
    int n = idx >> 7;
    float rs = rowsum2[n];
    rs = (rs == 0.f) ? 1e-12f : rs;
    out[idx] = kb_elu(H2[idx] / rs);
}

// ---------------------------------------------------------------------------
// Host launch
// ---------------------------------------------------------------------------
extern "C" void kernel_launch(void* const* d_in, const int* in_sizes, int n_in,
                              void* d_out, int out_size, void* d_ws, size_t ws_size,
                              hipStream_t stream)
{
    (void)in_sizes; (void)n_in; (void)out_size; (void)ws_size;
    const int N = 50000, E = 800000, EN = 200000, NREL = 500;

    const float* X          = (const float*)d_in[0];   // [N,128]
    const float* relE       = (const float*)d_in[1];   // [NREL,128]
    const int*   edge_list  = (const int*)d_in[2];     // [2,E]
    const int*   edge_type  = (const int*)d_in[3];     // [E]
    const float* edge_embed = (const float*)d_in[4];   // [E,128]
    const int*   el_nhop    = (const int*)d_in[5];     // [2,EN]
    const int*   et_nhop    = (const int*)d_in[6];     // [EN,2]
    const float* a_heads    = (const float*)d_in[7];   // [2,64,384]
    const float* a2_heads   = (const float*)d_in[8];   // [2,1,64]
    const float* W          = (const float*)d_in[9];   // [128,128]
    const float* a_out      = (const float*)d_in[10];  // [128,384]
    const float* a2_out     = (const float*)d_in[11];  // [1,128]

    float* ws = (float*)d_ws;
    size_t off = 0;
    auto alloc = [&](size_t n) { float* p = ws + off; off += n; return p; };
    float* R1       = alloc((size_t)E * 128);      // edge_embed @ C1
    float* P1       = alloc((size_t)N * 256);
    float* relC     = alloc((size_t)NREL * 128);
    float* H1       = alloc((size_t)N * 128);
    float* rowsum1  = alloc((size_t)N * 2);
    float* x2       = alloc((size_t)N * 128);
    float* AB1      = alloc(128 * 256);
    float* C1       = alloc(128 * 128);
    float* AB2      = alloc(128 * 256);
    float* Crel2    = alloc(128 * 128);
    float* P2       = alloc((size_t)N * 256);
    float* relproj2 = alloc((size_t)NREL * 128);
    float* H2       = alloc((size_t)N * 128);
    float* rowsum2  = alloc((size_t)N);

    // zero accumulators (graph-capture-safe async memsets)
    hipMemsetAsync(H1,      0, (size_t)N * 128 * sizeof(float), stream);
    hipMemsetAsync(rowsum1, 0, (size_t)N * 2   * sizeof(float), stream);
    hipMemsetAsync(H2,      0, (size_t)N * 128 * sizeof(float), stream);
    hipMemsetAsync(rowsum2, 0, (size_t)N       * sizeof(float), stream);

    // repack weights
    pack_AB1  <<<(128 * 256 + 255) / 256, 256, 0, stream>>>(a_heads, AB1);
    pack_C1   <<<(128 * 128 + 255) / 256, 256, 0, stream>>>(a_heads, C1);
    pack_AB2  <<<(128 * 256 + 255) / 256, 256, 0, stream>>>(a_out, AB2);
    pack_Crel2<<<(128 * 128 + 255) / 256, 256, 0, stream>>>(a_out, Crel2);

    auto gemm = [&](const float* A, const float* B, float* C, int M, int Nn) {
        int tiles = ((M + 15) / 16) * (Nn / 64);   // 16x64 tile per wave
        int grid = (tiles + 7) / 8;                // 8 waves per 256-thread block
        gemm_wmma_f32_k<128><<<grid, 256, 0, stream>>>(A, B, C, M, Nn);
    };

    // layer 1 projections
    gemm(X, AB1, P1, N, 256);
    gemm(relE, C1, relC, NREL, 128);
    gemm(edge_embed, C1, R1, E, 128);

    const int totE = E + EN;
    const int edgeBlocks = (totE + 7) / 8;         // wave per edge, 8 waves/block
    edge_att1<<<edgeBlocks, 256, 0, stream>>>(edge_list, el_nhop, et_nhop,
                                              P1, R1, relC, a2_heads,
                                              H1, rowsum1, E, EN);
    node1_finalize<<<(N * 128 + 255) / 256, 256, 0, stream>>>(H1, rowsum1, x2, N);

    // out_relation_1 -> second half of d_out
    float* out_rel = (float*)d_out + (size_t)N * 128;
    gemm(relE, W, out_rel, NREL, 128);
    gemm(out_rel, Crel2, relproj2, NREL, 128);
    gemm(x2, AB2, P2, N, 256);

    edge_att2<<<edgeBlocks, 256, 0, stream>>>(edge_list, el_nhop, edge_type, et_nhop,
                                              P2, relproj2, a2_out,
                                              H2, rowsum2, E, EN);
    node2_finalize<<<(N * 128 + 255) / 256, 256, 0, stream>>>(H2, rowsum2,
                                                              (float*)d_out, N);
}